// SDGL_SeqDe_part1_57466662421112
// MI455X (gfx1250) — compile-verified
//
#include <hip/hip_runtime.h>

// Problem constants (match reference): B=16, C=32, N=1024, T=64, C_OUT=32, ORDER=2
#define GB   16
#define GC   32
#define GN   1024
#define GT   64
#define TT   16               // t-tile width handled per workgroup (== WMMA N dim)
#define NBUF (GN * TT)        // floats per LDS activation buffer (1024x16 = 64KB)
#define SLICE ((size_t)GN * GT)        // elements per (b,c) slice = 65536

typedef __attribute__((ext_vector_type(2))) float v2f;
typedef __attribute__((ext_vector_type(8))) float v8f;

__device__ __forceinline__ v8f wmma_f32(v2f a, v2f b, v8f c) {
  // D = A(16x4) * B(4x16) + C, full fp32 (V_WMMA_F32_16X16X4_F32)
  return __builtin_amdgcn_wmma_f32_16x16x4_f32(false, a, false, b, (short)0, c,
                                               false, false);
}

// ---------------------------------------------------------------------------
// Kernel 1: per (b,c,t-tile) compute the 5-matmul graph-propagation chain:
//   y0 = AL @ x        ; s0 = x - y0
//   y1 = AL @ y0       ; s1 = AS @ s0 ; o1 = y1 + s1
//   y2 = AL @ y1       ; s2 = AS @ s1 ; o2 = y2 + s2
// Activation tiles [1024 x 16] live in LDS; AL/AS stream from L2.
// Grid: B*C*(T/TT) = 2048 blocks, 256 threads (8 waves, 8 node-tiles each).
// ---------------------------------------------------------------------------
__global__ __launch_bounds__(256) void sdgl_graph_prop(
    const float* __restrict__ x, const float* __restrict__ AL,
    const float* __restrict__ AS, float* __restrict__ o1,
    float* __restrict__ o2) {
  __shared__ float lds[4 * NBUF];  // 256 KB: x/y1 | s1 | y0 | s0
  float* bufX  = lds;              // holds x, later overwritten with y1
  float* bufS1 = lds + NBUF;
  float* bufY0 = lds + 2 * NBUF;
  float* bufS0 = lds + 3 * NBUF;

  const int tid  = threadIdx.x;
  const int lane = tid & 31;
  const int wave = tid >> 5;
  const int lrow = lane & 15;   // M (or N) position inside fragment
  const int lhi  = lane >> 4;   // half-wave select

  const int tblk = blockIdx.x & 3;
  const int bc   = blockIdx.x >> 2;
  const int t0   = tblk * TT;
  const size_t sliceOff = (size_t)bc * SLICE;

  // --- load x tile [1024 x 16] into bufX (coalesced float4) ---
  const float* xs = x + sliceOff + t0;
  for (int i = tid; i < GN * 4; i += 256) {
    const int n = i >> 2;
    const int q = (i & 3) * 4;
    *(float4*)(bufX + n * TT + q) = *(const float4*)(xs + (size_t)n * GT + q);
  }
  __syncthreads();

  const int i0base = wave * 128;  // 8 node-tiles of 16 rows per wave

  // ---- Stage 1: y0 = AL @ x ; s0 = x - y0 ----
  for (int r = 0; r < 8; ++r) {
    const int i0 = i0base + r * 16;
    v8f acc = {0.f, 0.f, 0.f, 0.f, 0.f, 0.f, 0.f, 0.f};
    const float* Arow = AL + (size_t)(i0 + lrow) * GN + 2 * lhi;
    const float* Bp   = bufX + (2 * lhi) * TT + lrow;
    for (int k = 0; k < GN; k += 4) {
      v2f a = *(const v2f*)(Arow + k);
      v2f b;
      b.x = Bp[k * TT];
      b.y = Bp[k * TT + TT];
      acc = wmma_f32(a, b, acc);
    }
#pragma unroll
    for (int v = 0; v < 8; ++v) {
      const int m = i0 + v + 8 * lhi;
      const float xv = bufX[m * TT + lrow];
      bufY0[m * TT + lrow] = acc[v];
      bufS0[m * TT + lrow] = xv - acc[v];
    }
  }
  __syncthreads();

  // ---- Stage 2: y1 = AL @ y0 -> bufX ; s1 = AS @ s0 -> bufS1 ; o1 out ----
  float* o1s = o1 + sliceOff + t0;
  for (int r = 0; r < 8; ++r) {
    const int i0 = i0base + r * 16;
    v8f accY = {0.f, 0.f, 0.f, 0.f, 0.f, 0.f, 0.f, 0.f};
    v8f accS = {0.f, 0.f, 0.f, 0.f, 0.f, 0.f, 0.f, 0.f};
    const float* ArL = AL + (size_t)(i0 + lrow) * GN + 2 * lhi;
    const float* ArS = AS + (size_t)(i0 + lrow) * GN + 2 * lhi;
    const float* BY  = bufY0 + (2 * lhi) * TT + lrow;
    const float* BS  = bufS0 + (2 * lhi) * TT + lrow;
    for (int k = 0; k < GN; k += 4) {
      v2f aL = *(const v2f*)(ArL + k);
      v2f bY;
      bY.x = BY[k * TT];
      bY.y = BY[k * TT + TT];
      accY = wmma_f32(aL, bY, accY);
      v2f aS = *(const v2f*)(ArS + k);
      v2f bS;
      bS.x = BS[k * TT];
      bS.y = BS[k * TT + TT];
      accS = wmma_f32(aS, bS, accS);
    }
#pragma unroll
    for (int v = 0; v < 8; ++v) {
      const int m = i0 + v + 8 * lhi;
      bufX[m * TT + lrow]  = accY[v];
      bufS1[m * TT + lrow] = accS[v];
      o1s[(size_t)m * GT + lrow] = accY[v] + accS[v];
    }
  }
  __syncthreads();

  // ---- Stage 3: o2 = AL @ y1 + AS @ s1 -> global only ----
  float* o2s = o2 + sliceOff + t0;
  for (int r = 0; r < 8; ++r) {
    const int i0 = i0base + r * 16;
    v8f accY = {0.f, 0.f, 0.f, 0.f, 0.f, 0.f, 0.f, 0.f};
    v8f accS = {0.f, 0.f, 0.f, 0.f, 0.f, 0.f, 0.f, 0.f};
    const float* ArL = AL + (size_t)(i0 + lrow) * GN + 2 * lhi;
    const float* ArS = AS + (size_t)(i0 + lrow) * GN + 2 * lhi;
    const float* BY  = bufX + (2 * lhi) * TT + lrow;
    const float* BS  = bufS1 + (2 * lhi) * TT + lrow;
    for (int k = 0; k < GN; k += 4) {
      v2f aL = *(const v2f*)(ArL + k);
      v2f bY;
      bY.x = BY[k * TT];
      bY.y = BY[k * TT + TT];
      accY = wmma_f32(aL, bY, accY);
      v2f aS = *(const v2f*)(ArS + k);
      v2f bS;
      bS.x = BS[k * TT];
      bS.y = BS[k * TT + TT];
      accS = wmma_f32(aS, bS, accS);
    }
#pragma unroll
    for (int v = 0; v < 8; ++v) {
      const int m = i0 + v + 8 * lhi;
      o2s[(size_t)m * GT + lrow] = accY[v] + accS[v];
    }
  }
}

// ---------------------------------------------------------------------------
// Kernel 2: 1x1 conv channel mix via WMMA.
//   out[b,o,p] = bias[o] + sum_{cc<96} W[o,cc] * H[cc,p],
//   H rows 0..31 = x, 32..63 = o1, 64..95 = o2 (each row = contiguous 65536).
// Grid: B * 512 blocks, 256 threads; each wave does one 16-col tile, M=32.
// ---------------------------------------------------------------------------
__global__ __launch_bounds__(256) void sdgl_channel_mix(
    const float* __restrict__ x, const float* __restrict__ o1,
    const float* __restrict__ o2, const float* __restrict__ W,
    const float* __restrict__ bias, float* __restrict__ out) {
  const int tid  = threadIdx.x;
  const int lane = tid & 31;
  const int wave = tid >> 5;
  const int lrow = lane & 15;
  const int lhi  = lane >> 4;

  const int b    = blockIdx.x >> 9;                 // 512 blocks per batch
  const int pblk = ((blockIdx.x & 511) << 3) + wave;  // 4096 col-tiles per batch
  const size_t p0 = (size_t)pblk * 16;

  const size_t bOff = (size_t)b * GC * SLICE;
  const float* srcs[3] = {x + bOff, o1 + bOff, o2 + bOff};

  v8f acc0, acc1;
#pragma unroll
  for (int v = 0; v < 8; ++v) {
    acc0[v] = bias[v + 8 * lhi];
    acc1[v] = bias[16 + v + 8 * lhi];
  }

  for (int g = 0; g < 3; ++g) {
    const float* H = srcs[g];
    for (int kc = 0; kc < GC; kc += 4) {
      const int cc = g * GC + kc;
      v2f a0 = *(const v2f*)(W + (size_t)lrow * 96 + cc + 2 * lhi);
      v2f a1 = *(const v2f*)(W + (size_t)(16 + lrow) * 96 + cc + 2 * lhi);
      v2f bf;
      bf.x = H[(size_t)(kc + 2 * lhi) * SLICE + p0 + lrow];
      bf.y = H[(size_t)(kc + 2 * lhi + 1) * SLICE + p0 + lrow];
      acc0 = wmma_f32(a0, bf, acc0);
      acc1 = wmma_f32(a1, bf, acc1);
    }
  }

  float* ob = out + (size_t)b * 32 * SLICE;
#pragma unroll
  for (int v = 0; v < 8; ++v) {
    ob[(size_t)(v + 8 * lhi) * SLICE + p0 + lrow]      = acc0[v];
    ob[(size_t)(16 + v + 8 * lhi) * SLICE + p0 + lrow] = acc1[v];
  }
}

extern "C" void kernel_launch(void* const* d_in, const int* in_sizes, int n_in,
                              void* d_out, int out_size, void* d_ws,
                              size_t ws_size, hipStream_t stream) {
  const float* x    = (const float*)d_in[0];
  const float* AL   = (const float*)d_in[1];
  const float* AS   = (const float*)d_in[2];
  const float* W    = (const float*)d_in[3];
  const float* bias = (const float*)d_in[4];
  float* out = (float*)d_out;

  // workspace: o1 and o2, each [B,C,N,T] fp32 (134 MB each)
  float* o1 = (float*)d_ws;
  float* o2 = o1 + (size_t)GB * GC * SLICE;

  sdgl_graph_prop<<<GB * GC * (GT / TT), 256, 0, stream>>>(x, AL, AS, o1, o2);
  sdgl_channel_mix<<<GB * 512, 256, 0, stream>>>(x, o1, o2, W, bias, out);
}